// SlotMemoryModel_3204045603555
// MI455X (gfx1250) — compile-verified
//
#include <hip/hip_runtime.h>
#include <hip/hip_bf16.h>

#define HD 64      // hidden dim
#define VS 128     // vocab size
#define NS 64      // num slots
#define BL 4096    // sequence length

typedef __attribute__((ext_vector_type(2))) float v2f;
typedef __attribute__((ext_vector_type(8))) float v8f;

// One workgroup (8 wave32) per batch element. All GEMMs on V_WMMA_F32_16X16X4_F32.
// Loop-invariant weight B-fragments live in VGPRs; einsum accumulator lives in VGPRs.
__global__ __launch_bounds__(256)
void slot_memory_fused(const int* __restrict__ seq,
                       const float* __restrict__ embed_w,
                       const float* __restrict__ w1,
                       const float* __restrict__ b1,
                       const float* __restrict__ w2,
                       const float* __restrict__ b2,
                       const float* __restrict__ ln_g,
                       const float* __restrict__ ln_b,
                       const float* __restrict__ slot_keys,
                       const float* __restrict__ gate_w,
                       const float* __restrict__ gate_b,
                       const float* __restrict__ out_w,
                       const float* __restrict__ out_b,
                       float* __restrict__ out)
{
    // ---- LDS (~150.5 KB of 320 KB/WGP) ----
    __shared__ float s_embed[VS * HD];      // 32 KB
    __shared__ float s_w1[HD * 2 * HD];     // 32 KB
    __shared__ float s_w2[2 * HD * HD];     // 32 KB
    __shared__ float s_gw[HD * NS];         // 16 KB
    __shared__ float s_b1[2 * HD];
    __shared__ float s_b2[HD];
    __shared__ float s_lng[HD];
    __shared__ float s_lnb[HD];
    __shared__ float s_gb[NS];
    __shared__ float s_h0[16 * HD];         // 4 KB  pre-LN embeddings for tile
    __shared__ float s_r[16 * 2 * HD];      // 8 KB  relu(h@w1+b1)
    __shared__ float s_h[16 * HD];          // 4 KB  post-LN hidden
    __shared__ float s_g[16 * NS];          // 4 KB  gate logits -> softmax weights
    __shared__ float s_keys[NS * HD];       // 16 KB final keys (= vals)
    __shared__ float s_q[HD];
    __shared__ float s_sim[NS];
    __shared__ float s_attn[NS];
    __shared__ float s_ctx[HD];

    const int tid   = threadIdx.x;
    const int b     = blockIdx.x;
    const int lane  = tid & 31;
    const int wv    = tid >> 5;             // wave id 0..7
    const int lrow  = lane & 15;            // M-row (A) / N-col (B,C,D)
    const int khalf = (lane >> 4) << 1;     // K offset per lane-half {0,2}

    // ---- stage 0: weights -> LDS ----
    for (int i = tid; i < VS * HD;     i += 256) s_embed[i] = embed_w[i];
    for (int i = tid; i < HD * 2 * HD; i += 256) s_w1[i]    = w1[i];
    for (int i = tid; i < 2 * HD * HD; i += 256) s_w2[i]    = w2[i];
    for (int i = tid; i < HD * NS;     i += 256) s_gw[i]    = gate_w[i];
    if (tid < 2 * HD) s_b1[tid] = b1[tid];
    if (tid < HD) { s_b2[tid] = b2[tid]; s_lng[tid] = ln_g[tid]; s_lnb[tid] = ln_b[tid]; }
    if (tid < NS) s_gb[tid] = gate_b[tid];
    __syncthreads();

    const int* seqb = seq + (size_t)b * BL;

    // ---- hoist loop-invariant B-operand fragments into VGPRs ----
    // GEMM1 (all waves): w1 columns [wv*16, wv*16+16), K=64 -> 16 x v2f
    v2f fw1[16];
#pragma unroll
    for (int kk = 0; kk < 16; ++kk) {
        int k4 = kk * 4 + khalf;
        fw1[kk].x = s_w1[(k4)     * (2 * HD) + wv * 16 + lrow];
        fw1[kk].y = s_w1[(k4 + 1) * (2 * HD) + wv * 16 + lrow];
    }
    // Overlaid fragment array: waves 0..3 -> w2 (K=128, 32 frags);
    // waves 4..7 -> gate_w (K=64, 16 frags). Wave-uniform branches share registers.
    v2f fbig[32];
    if (wv < 4) {
        const int n0 = wv * 16;
#pragma unroll
        for (int kk = 0; kk < 32; ++kk) {
            int k4 = kk * 4 + khalf;
            fbig[kk].x = s_w2[(k4)     * HD + n0 + lrow];
            fbig[kk].y = s_w2[(k4 + 1) * HD + n0 + lrow];
        }
    } else {
        const int n0 = (wv - 4) * 16;
#pragma unroll
        for (int kk = 0; kk < 16; ++kk) {
            int k4 = kk * 4 + khalf;
            fbig[kk].x = s_gw[(k4)     * NS + n0 + lrow];
            fbig[kk].y = s_gw[(k4 + 1) * NS + n0 + lrow];
        }
#pragma unroll
        for (int kk = 16; kk < 32; ++kk) { fbig[kk].x = 0.f; fbig[kk].y = 0.f; }
    }

    // einsum accumulator: 64x64 = 16 WMMA tiles, 2 per wave, VGPR-resident
    v8f acc0 = {};
    v8f acc1 = {};
    const int t0  = wv * 2;
    const int m0a = (t0 >> 2) * 16,        n0a = (t0 & 3) * 16;
    const int m0b = ((t0 + 1) >> 2) * 16,  n0b = ((t0 + 1) & 3) * 16;

    for (int tile = 0; tile < BL / 16; ++tile) {
        // ---- A: embedding gather for 16 tokens (+ prefetch next tile's ids) ----
        {
            int row = tid >> 4;
            int c4  = (tid & 15) * 4;
            int tok = seqb[tile * 16 + row];
            const float* e = &s_embed[tok * HD + c4];
            s_h0[row * HD + c4 + 0] = e[0];
            s_h0[row * HD + c4 + 1] = e[1];
            s_h0[row * HD + c4 + 2] = e[2];
            s_h0[row * HD + c4 + 3] = e[3];
            if (tid == 0 && tile + 1 < BL / 16)
                __builtin_prefetch(&seqb[(tile + 1) * 16], 0, 0);
        }
        __syncthreads();

        // ---- B: r = relu(h0 @ w1 + b1), 16x128, one 16x16 tile per wave ----
        {
            v8f c = {};
#pragma unroll
            for (int kk = 0; kk < 16; ++kk) {
                v2f a = *(const v2f*)&s_h0[lrow * HD + kk * 4 + khalf];
                c = __builtin_amdgcn_wmma_f32_16x16x4_f32(false, a, false, fw1[kk], (short)0, c, false, false);
            }
            const int n  = wv * 16 + lrow;
            const int mb = (lane >> 4) * 8;
            const float bias = s_b1[n];
#pragma unroll
            for (int v = 0; v < 8; ++v)
                s_r[(mb + v) * (2 * HD) + n] = fmaxf(c[v] + bias, 0.f);
        }
        __syncthreads();

        // ---- C: h1 = h0 + (r @ w2 + b2), 16x64, waves 0..3 ----
        if (wv < 4) {
            v8f c = {};
#pragma unroll
            for (int kk = 0; kk < 32; ++kk) {
                v2f a = *(const v2f*)&s_r[lrow * (2 * HD) + kk * 4 + khalf];
                c = __builtin_amdgcn_wmma_f32_16x16x4_f32(false, a, false, fbig[kk], (short)0, c, false, false);
            }
            const int n  = wv * 16 + lrow;
            const int mb = (lane >> 4) * 8;
            const float bias = s_b2[n];
#pragma unroll
            for (int v = 0; v < 8; ++v) {
                int m = mb + v;
                s_h[m * HD + n] = s_h0[m * HD + n] + c[v] + bias;
            }
        }
        __syncthreads();

        // ---- D: LayerNorm, wave-parallel: 2 rows/wave, 16 lanes/row, 4 elems/lane ----
        {
            const int row = 2 * wv + (lane >> 4);
            const int c4  = (lane & 15) * 4;
            float* hr = &s_h[row * HD + c4];
            float x0 = hr[0], x1 = hr[1], x2 = hr[2], x3 = hr[3];
            float s = x0 + x1 + x2 + x3;
            s += __shfl_xor(s, 1); s += __shfl_xor(s, 2);
            s += __shfl_xor(s, 4); s += __shfl_xor(s, 8);
            float mu = s * (1.f / HD);
            float d0 = x0 - mu, d1 = x1 - mu, d2 = x2 - mu, d3 = x3 - mu;
            float vs = d0 * d0 + d1 * d1 + d2 * d2 + d3 * d3;
            vs += __shfl_xor(vs, 1); vs += __shfl_xor(vs, 2);
            vs += __shfl_xor(vs, 4); vs += __shfl_xor(vs, 8);
            float rs = rsqrtf(vs * (1.f / HD) + 1e-5f);
            float y0 = d0 * rs * s_lng[c4 + 0] + s_lnb[c4 + 0];
            float y1 = d1 * rs * s_lng[c4 + 1] + s_lnb[c4 + 1];
            float y2 = d2 * rs * s_lng[c4 + 2] + s_lnb[c4 + 2];
            float y3 = d3 * rs * s_lng[c4 + 3] + s_lnb[c4 + 3];
            hr[0] = y0; hr[1] = y1; hr[2] = y2; hr[3] = y3;
            if (tile * 16 + row == BL - 1) {   // capture query h[L-1]
                s_q[c4 + 0] = y0; s_q[c4 + 1] = y1;
                s_q[c4 + 2] = y2; s_q[c4 + 3] = y3;
            }
        }
        __syncthreads();

        // ---- E: gate logits g = h @ gate_w, 16x64, waves 4..7 ----
        if (wv >= 4) {
            v8f c = {};
#pragma unroll
            for (int kk = 0; kk < 16; ++kk) {
                v2f a = *(const v2f*)&s_h[lrow * HD + kk * 4 + khalf];
                c = __builtin_amdgcn_wmma_f32_16x16x4_f32(false, a, false, fbig[kk], (short)0, c, false, false);
            }
            const int n  = (wv - 4) * 16 + lrow;
            const int mb = (lane >> 4) * 8;
#pragma unroll
            for (int v = 0; v < 8; ++v)
                s_g[(mb + v) * NS + n] = c[v];
        }
        __syncthreads();

        // ---- F: row softmax (+gate_b), wave-parallel; zero the last-token row ----
        {
            const int row = 2 * wv + (lane >> 4);
            const int c4  = (lane & 15) * 4;
            float* gr = &s_g[row * NS + c4];
            const bool lastTok = (tile * 16 + row == BL - 1);
            float t0v = gr[0] + s_gb[c4 + 0];
            float t1v = gr[1] + s_gb[c4 + 1];
            float t2v = gr[2] + s_gb[c4 + 2];
            float t3v = gr[3] + s_gb[c4 + 3];
            float mx = fmaxf(fmaxf(t0v, t1v), fmaxf(t2v, t3v));
            mx = fmaxf(mx, __shfl_xor(mx, 1)); mx = fmaxf(mx, __shfl_xor(mx, 2));
            mx = fmaxf(mx, __shfl_xor(mx, 4)); mx = fmaxf(mx, __shfl_xor(mx, 8));
            float e0 = __expf(t0v - mx), e1 = __expf(t1v - mx);
            float e2 = __expf(t2v - mx), e3 = __expf(t3v - mx);
            float sum = e0 + e1 + e2 + e3;
            sum += __shfl_xor(sum, 1); sum += __shfl_xor(sum, 2);
            sum += __shfl_xor(sum, 4); sum += __shfl_xor(sum, 8);
            float inv = lastTok ? 0.f : (1.f / sum);  // exclude query token from hw
            gr[0] = e0 * inv; gr[1] = e1 * inv;
            gr[2] = e2 * inv; gr[3] = e3 * inv;
        }
        __syncthreads();

        // ---- G: acc(64x64) += w_tile^T(64x16) @ h(16x64); 2 tiles/wave, VGPR accum ----
#pragma unroll
        for (int k = 0; k < 16; k += 4) {
            v2f a0, b0, a1, b1v;
            a0.x  = s_g[(k + khalf)     * NS + m0a + lrow];   // A[m][kk] = w[kk][m]
            a0.y  = s_g[(k + khalf + 1) * NS + m0a + lrow];
            b0.x  = s_h[(k + khalf)     * HD + n0a + lrow];   // B[kk][n] = h[kk][n]
            b0.y  = s_h[(k + khalf + 1) * HD + n0a + lrow];
            acc0 = __builtin_amdgcn_wmma_f32_16x16x4_f32(false, a0, false, b0, (short)0, acc0, false, false);
            a1.x  = s_g[(k + khalf)     * NS + m0b + lrow];
            a1.y  = s_g[(k + khalf + 1) * NS + m0b + lrow];
            b1v.x = s_h[(k + khalf)     * HD + n0b + lrow];
            b1v.y = s_h[(k + khalf + 1) * HD + n0b + lrow];
            acc1 = __builtin_amdgcn_wmma_f32_16x16x4_f32(false, a1, false, b1v, (short)0, acc1, false, false);
        }
        __syncthreads();
    }

    // ---- finale: keys = slot_keys + acc ----
    {
        const int mb = (lane >> 4) * 8;
#pragma unroll
        for (int v = 0; v < 8; ++v) {
            int m = m0a + mb + v, n = n0a + lrow;
            s_keys[m * HD + n] = acc0[v] + slot_keys[m * HD + n];
        }
#pragma unroll
        for (int v = 0; v < 8; ++v) {
            int m = m0b + mb + v, n = n0b + lrow;
            s_keys[m * HD + n] = acc1[v] + slot_keys[m * HD + n];
        }
    }
    __syncthreads();

    // cosine similarity sim[n] = <keys[n]/||keys[n]||, q/||q||>
    if (tid < NS) {
        float qn = 0.f;
        for (int i = 0; i < HD; ++i) qn += s_q[i] * s_q[i];
        qn = fmaxf(sqrtf(qn), 1e-12f);
        const float* kr = &s_keys[tid * HD];
        float dot = 0.f, kn = 0.f;
        for (int i = 0; i < HD; ++i) { float kv = kr[i]; dot += kv * s_q[i]; kn += kv * kv; }
        kn = fmaxf(sqrtf(kn), 1e-12f);
        s_sim[tid] = dot / (kn * qn);
    }
    __syncthreads();

    // softmax over slots
    if (tid < NS) {
        float mx = -3.4e38f;
        for (int i = 0; i < NS; ++i) mx = fmaxf(mx, s_sim[i]);
        float sum = 0.f;
        for (int i = 0; i < NS; ++i) sum += __expf(s_sim[i] - mx);
        s_attn[tid] = __expf(s_sim[tid] - mx) / sum;
    }
    __syncthreads();

    // ctx[h] = sum_n attn[n] * keys[n][h]   (vals == keys)
    if (tid < HD) {
        float c = 0.f;
        for (int n = 0; n < NS; ++n) c += s_attn[n] * s_keys[n * HD + tid];
        s_ctx[tid] = c;
    }
    __syncthreads();

    // out[b][v] = ctx @ out_w + out_b
    if (tid < VS) {
        float o = out_b[tid];
        for (int h = 0; h < HD; ++h) o += s_ctx[h] * out_w[h * VS + tid];
        out[(size_t)b * VS + tid] = o;
    }
}

extern "C" void kernel_launch(void* const* d_in, const int* in_sizes, int n_in,
                              void* d_out, int out_size, void* d_ws, size_t ws_size,
                              hipStream_t stream) {
    (void)in_sizes; (void)n_in; (void)out_size; (void)d_ws; (void)ws_size;
    const int*   seq       = (const int*)  d_in[0];
    const float* embed_w   = (const float*)d_in[1];
    const float* w1        = (const float*)d_in[2];
    const float* b1        = (const float*)d_in[3];
    const float* w2        = (const float*)d_in[4];
    const float* b2        = (const float*)d_in[5];
    const float* ln_g      = (const float*)d_in[6];
    const float* ln_b      = (const float*)d_in[7];
    const float* slot_keys = (const float*)d_in[8];
    // d_in[9] = slot_vals: unused (reference sets vals = keys)
    const float* gate_w    = (const float*)d_in[10];
    const float* gate_b    = (const float*)d_in[11];
    const float* out_w     = (const float*)d_in[12];
    const float* out_b     = (const float*)d_in[13];
    float* out = (float*)d_out;

    slot_memory_fused<<<dim3(256), dim3(256), 0, stream>>>(
        seq, embed_w, w1, b1, w2, b2, ln_g, ln_b,
        slot_keys, gate_w, gate_b, out_w, out_b, out);
}